// CostBuilder_22488448761897
// MI455X (gfx1250) — compile-verified
//
#include <hip/hip_runtime.h>
#include <stdint.h>

// Problem constants from the reference: (B,C,H,W)=(4,32,64,128), D=192/4=48.
#define Bsz 4
#define Csz 32
#define Hsz 64
#define Wsz 128
#define Dsz 48
#define HWsz (Hsz * Wsz)

typedef __attribute__((ext_vector_type(4))) unsigned int v4u;
typedef __attribute__((ext_vector_type(8))) int          v8i;
typedef __attribute__((ext_vector_type(4))) int          v4i;
typedef __attribute__((ext_vector_type(4))) float        f4;

#if defined(__gfx1250__) && __has_builtin(__builtin_amdgcn_tensor_load_to_lds)
#define HAVE_TDM 1
#else
#define HAVE_TDM 0
#endif

#if HAVE_TDM
// Issue one TDM row load: 128 x f32 (512B) tile from global -> LDS.
// D# per CDNA5 ISA ch.8: group0 {count=1, lds_addr, global_addr, type=2},
// group1 {data_size=4B, tensor_dim0=128, tile_dim0=128, tile_dim1=1, strides=128}.
// This toolchain exposes the 6-arg builtin: (g0, g1, g2, g3, extra, cpol).
__device__ __forceinline__ void tdm_load_row(const float* gptr, uint32_t lds_off) {
  uint64_t ga = (uint64_t)(uintptr_t)gptr;
  v4u g0;
  g0[0] = 1u;                                                 // count=1, no gather
  g0[1] = lds_off;                                            // lds_addr (bytes)
  g0[2] = (uint32_t)ga;                                       // global_addr[31:0]
  g0[3] = ((uint32_t)(ga >> 32) & 0x01FFFFFFu) | 0x80000000u; // addr[56:32] | type=2
  v8i g1;
  g1[0] = (int)(2u << 16);                // data_size=2 (4 bytes); wg_mask=0
  g1[1] = (int)((uint32_t)Wsz << 16);     // tensor_dim0[15:0] = 128
  g1[2] = (int)(1u << 16);                // tensor_dim0 hi=0 | tensor_dim1 lo = 1
  g1[3] = (int)((uint32_t)Wsz << 16);     // tensor_dim1 hi=0 | tile_dim0 = 128
  g1[4] = 1;                              // tile_dim1 = 1, tile_dim2 = 0
  g1[5] = Wsz;                            // tensor_dim0_stride lo32 = 128
  g1[6] = (int)((uint32_t)Wsz << 16);     // stride0 hi=0 | tensor_dim1_stride lo16
  g1[7] = 0;                              // tensor_dim1_stride hi
  v4i z4 = {0, 0, 0, 0};
  v8i z8 = {0, 0, 0, 0, 0, 0, 0, 0};
  __builtin_amdgcn_tensor_load_to_lds(g0, g1, z4, z4, z8, 0);
}
#endif

__global__ __launch_bounds__(256) void CostBuilder_kernel(
    const float* __restrict__ left, const float* __restrict__ right,
    float* __restrict__ out) {
  __shared__ float sL[Wsz];
  __shared__ float sR[Wsz];

  const int blk = blockIdx.x;          // one block per (b,c,h)
  const int h = blk % Hsz;
  const int c = (blk / Hsz) % Csz;
  const int b = blk / (Hsz * Csz);

  const size_t row_off = ((size_t)((b * Csz + c) * Hsz + h)) * Wsz;

#if HAVE_TDM
  if (threadIdx.x < 32) {              // wave 0 drives the DMA engine
    tdm_load_row(left + row_off, (uint32_t)(uintptr_t)&sL[0]);
    tdm_load_row(right + row_off, (uint32_t)(uintptr_t)&sR[0]);
    __builtin_amdgcn_s_wait_tensorcnt(0);
  }
#else
  if (threadIdx.x < 32) {              // fallback: 32 lanes x float4 = 128 floats
    ((f4*)sL)[threadIdx.x] = ((const f4*)(left + row_off))[threadIdx.x];
    ((f4*)sR)[threadIdx.x] = ((const f4*)(right + row_off))[threadIdx.x];
  }
#endif
  __syncthreads();

  const int tid = threadIdx.x;
  const int part = tid >> 7;           // 0: left half of channels, 1: right half
  const int lane4 = tid & 31;          // which float4 of the 128-wide row
  const int rowslot = (tid >> 5) & 3;  // 4 row-slots -> d = rowslot + 4*i
  const int w0 = lane4 * 4;

  const size_t outBase =
      ((size_t)(b * (2 * Csz) + part * Csz + c)) * ((size_t)Dsz * HWsz) +
      (size_t)h * Wsz;

  if (part == 0) {
    // Left part: row is invariant in d; hoist into registers, mask per d.
    f4 lv;
    lv.x = sL[w0 + 0]; lv.y = sL[w0 + 1]; lv.z = sL[w0 + 2]; lv.w = sL[w0 + 3];
#pragma unroll
    for (int i = 0; i < 12; ++i) {
      const int d = rowslot + 4 * i;   // covers d = 0..47
      f4 v;
      v.x = (w0 + 0 >= d) ? lv.x : 0.0f;
      v.y = (w0 + 1 >= d) ? lv.y : 0.0f;
      v.z = (w0 + 2 >= d) ? lv.z : 0.0f;
      v.w = (w0 + 3 >= d) ? lv.w : 0.0f;
      __builtin_nontemporal_store(
          v, (f4*)(out + outBase + (size_t)d * HWsz + w0));
    }
  } else {
    // Right part: shifted gather right[w-d] from LDS, clamped like the
    // reference's clip(w-d, 0, W-1); mask zeroes w<d anyway.
#pragma unroll
    for (int i = 0; i < 12; ++i) {
      const int d = rowslot + 4 * i;
      f4 v;
      const int j0 = w0 + 0 - d, j1 = w0 + 1 - d, j2 = w0 + 2 - d, j3 = w0 + 3 - d;
      const float r0 = sR[j0 < 0 ? 0 : j0];
      const float r1 = sR[j1 < 0 ? 0 : j1];
      const float r2 = sR[j2 < 0 ? 0 : j2];
      const float r3 = sR[j3 < 0 ? 0 : j3];
      v.x = (j0 >= 0) ? r0 : 0.0f;
      v.y = (j1 >= 0) ? r1 : 0.0f;
      v.z = (j2 >= 0) ? r2 : 0.0f;
      v.w = (j3 >= 0) ? r3 : 0.0f;
      __builtin_nontemporal_store(
          v, (f4*)(out + outBase + (size_t)d * HWsz + w0));
    }
  }
}

extern "C" void kernel_launch(void* const* d_in, const int* in_sizes, int n_in,
                              void* d_out, int out_size, void* d_ws, size_t ws_size,
                              hipStream_t stream) {
  const float* left  = (const float*)d_in[0];
  const float* right = (const float*)d_in[1];
  float* out = (float*)d_out;
  (void)in_sizes; (void)n_in; (void)out_size; (void)d_ws; (void)ws_size;

  dim3 grid(Bsz * Csz * Hsz);  // 8192 blocks, one per (b,c,h)
  dim3 block(256);             // 8 wave32s
  CostBuilder_kernel<<<grid, block, 0, stream>>>(left, right, out);
}